// WaveletDecompose_2680059593339
// MI455X (gfx1250) — compile-verified
//
#include <hip/hip_runtime.h>

typedef __attribute__((ext_vector_type(2))) float v2f;
typedef __attribute__((ext_vector_type(8))) float v8f;

#define NIMG 16

// ---------------------------------------------------------------------------
// Compile-time A-fragment table for the banded DWT filter matrices.
// F[f][i][t] = f_rev[t - 2i] if 0 <= t-2i <= 7 else 0, f_rev[j] = DEC[f][7-j].
// Stored directly in WMMA 16x16x4 f32 A-fragment layout:
//   element (M = lane&15, K = 4*kb + v + 2*(lane>>4)), K padded to 40.
// ---------------------------------------------------------------------------
struct AFragTab {
    float v[2][10][2][32];
    constexpr AFragTab() : v{} {
        const float lo[8] = {-0.010597401784997278f, 0.032883011666982945f,
                              0.030841381835986965f, -0.18703481171888114f,
                             -0.02798376941698385f,  0.6308807679295904f,
                              0.7148465705525415f,   0.23037781330885523f};
        const float hi[8] = {-0.23037781330885523f,  0.7148465705525415f,
                             -0.6308807679295904f,  -0.02798376941698385f,
                              0.18703481171888114f,  0.030841381835986965f,
                             -0.032883011666982945f, -0.010597401784997278f};
        for (int f = 0; f < 2; ++f)
            for (int kb = 0; kb < 10; ++kb)
                for (int vv = 0; vv < 2; ++vv)
                    for (int lane = 0; lane < 32; ++lane) {
                        int M = lane & 15;
                        int h = lane >> 4;
                        int K = 4 * kb + vv + 2 * h;
                        int j = K - 2 * M;
                        float val = 0.0f;
                        if (j >= 0 && j <= 7)
                            val = (f == 0) ? lo[7 - j] : hi[7 - j];
                        v[f][kb][vv][lane] = val;
                    }
    }
};
__constant__ AFragTab AFRAG;   // constexpr default ctor -> constant-initialized

__device__ __forceinline__ int symidx(int i, int n) {
    if (i < 0)  i = -1 - i;          // symmetric (half-sample) extension
    if (i >= n) i = 2 * n - 1 - i;
    if (i < 0)  i = 0;               // safety clamp (only feeds garbage outputs)
    if (i >= n) i = n - 1;
    return i;
}

// One wave32 computes a 16x16 output tile of all four subbands for one image.
// Out_{b,a} = F_b * X(38x38 extended tile) * F_a^T, evaluated as two WMMA
// passes with the 38x38 tile (and the 16x48 intermediate) staged in LDS.
__global__ __launch_bounds__(32)
void dwt2_wmma_kernel(const float* __restrict__ src, int n, int m,
                      float* __restrict__ outA, float* __restrict__ outH,
                      float* __restrict__ outV, float* __restrict__ outD)
{
    __shared__ float Xs[48 * 41];       // 38x38 valid, zero-padded; stride 41 (bank-conflict-free)
    __shared__ float Us[2][16 * 49];    // U_f = F_f * X^T, 16x48; stride 49

    const int lane = threadIdx.x;       // 0..31
    const int nl   = lane & 15;
    const int h    = lane >> 4;
    const int r0   = blockIdx.y * 16;
    const int c0   = blockIdx.x * 16;
    const int img  = blockIdx.z;
    const size_t ibase = (size_t)img * (size_t)n * (size_t)n;

    // Phase 0: zero LDS tile (provides K/N zero padding for WMMA)
    for (int i = lane; i < 48 * 41; i += 32) Xs[i] = 0.0f;
    __syncthreads();

    // Phase 1: gather symmetric-extended 38x38 input tile
    for (int i = lane; i < 38 * 38; i += 32) {
        int tr = i / 38;
        int tc = i - tr * 38;
        int gr = symidx(2 * r0 - 6 + tr, n);
        int gc = symidx(2 * c0 - 6 + tc, n);
        Xs[tr * 41 + tc] = src[ibase + (size_t)gr * n + gc];
    }
    __syncthreads();

    // Phase 2 (row pass): U_f = F_f * X^T  (16 x 48), 3 chunks of N=16, K=40
    for (int f = 0; f < 2; ++f) {
        for (int c = 0; c < 3; ++c) {
            v8f acc = {};
            #pragma unroll
            for (int kb = 0; kb < 10; ++kb) {
                v2f a, b;
                a.x = AFRAG.v[f][kb][0][lane];
                a.y = AFRAG.v[f][kb][1][lane];
                // B element (k, n_local) = X^T[k][c*16+nl] = X[c*16+nl][k]
                int xr = (c * 16 + nl) * 41 + 4 * kb + 2 * h;
                b.x = Xs[xr];
                b.y = Xs[xr + 1];
                acc = __builtin_amdgcn_wmma_f32_16x16x4_f32(
                          false, a, false, b, (short)0, acc, false, false);
            }
            // D element (M = vv + 8h, N = nl) -> Us[f][M][c*16+nl]
            #pragma unroll
            for (int vv = 0; vv < 8; ++vv)
                Us[f][(vv + 8 * h) * 49 + c * 16 + nl] = acc[vv];
        }
    }
    __syncthreads();

    // Phase 3 (column pass): Out_{fb,fa} = F_fb * U_fa^T  (16 x 16), K=40
    const size_t obase = (size_t)img * (size_t)m * (size_t)m;
    for (int fb = 0; fb < 2; ++fb) {
        for (int fa = 0; fa < 2; ++fa) {
            v8f acc = {};
            #pragma unroll
            for (int kb = 0; kb < 10; ++kb) {
                v2f a, b;
                a.x = AFRAG.v[fb][kb][0][lane];
                a.y = AFRAG.v[fb][kb][1][lane];
                // B element (k, n_local) = U_fa^T[k][nl] = U_fa[nl][k]
                int ur = nl * 49 + 4 * kb + 2 * h;
                b.x = Us[fa][ur];
                b.y = Us[fa][ur + 1];
                acc = __builtin_amdgcn_wmma_f32_16x16x4_f32(
                          false, a, false, b, (short)0, acc, false, false);
            }
            // cA=(lo,lo)  cH=(hi col, lo row)  cV=(lo col, hi row)  cD=(hi,hi)
            float* dst = (fb == 0) ? (fa == 0 ? outA : outV)
                                   : (fa == 0 ? outH : outD);
            int cc = c0 + nl;
            if (cc < m) {
                #pragma unroll
                for (int vv = 0; vv < 8; ++vv) {
                    int rr = r0 + vv + 8 * h;
                    if (rr < m)
                        dst[obase + (size_t)rr * m + cc] = acc[vv];
                }
            }
        }
    }
}

static inline void launch_level(const float* src, int n, int m,
                                float* oA, float* oH, float* oV, float* oD,
                                hipStream_t stream)
{
    int tiles = (m + 15) / 16;
    dim3 grid(tiles, tiles, NIMG), block(32);
    hipLaunchKernelGGL(dwt2_wmma_kernel, grid, block, 0, stream,
                       src, n, m, oA, oH, oV, oD);
}

extern "C" void kernel_launch(void* const* d_in, const int* in_sizes, int n_in,
                              void* d_out, int out_size, void* d_ws, size_t ws_size,
                              hipStream_t stream)
{
    (void)in_sizes; (void)n_in; (void)out_size; (void)ws_size;
    const float* x = (const float*)d_in[0];   // (16,1,1024,1024) fp32
    float* out = (float*)d_out;

    const int m1 = 515, m2 = 261, m3 = 134;   // (n+7)//2 per level
    const size_t s1 = (size_t)m1 * m1, s2 = (size_t)m2 * m2, s3 = (size_t)m3 * m3;

    // Workspace: cA1 (16*515^2) + cA2 (16*261^2) floats (~20.4 MB)
    float* a1 = (float*)d_ws;
    float* a2 = a1 + (size_t)NIMG * s1;

    // d_out flat layout: cA3, H3, V3, D3, H2, V2, D2, H1, V1, D1
    float* oA3 = out;
    float* oH3 = oA3 + (size_t)NIMG * s3;
    float* oV3 = oH3 + (size_t)NIMG * s3;
    float* oD3 = oV3 + (size_t)NIMG * s3;
    float* oH2 = oD3 + (size_t)NIMG * s3;
    float* oV2 = oH2 + (size_t)NIMG * s2;
    float* oD2 = oV2 + (size_t)NIMG * s2;
    float* oH1 = oD2 + (size_t)NIMG * s2;
    float* oV1 = oH1 + (size_t)NIMG * s1;
    float* oD1 = oV1 + (size_t)NIMG * s1;

    launch_level(x,  1024, m1, a1,  oH1, oV1, oD1, stream);
    launch_level(a1, m1,   m2, a2,  oH2, oV2, oD2, stream);
    launch_level(a2, m2,   m3, oA3, oH3, oV3, oD3, stream);
}